// hSim_50732153700874
// MI455X (gfx1250) — compile-verified
//
#include <hip/hip_runtime.h>
#include <math.h>

// ---------------------------------------------------------------------------
// Fused  log_softmax( (X@U) @ (MU@V)^T )  for B=32, D=128, K=512, C=10.
// Single workgroup (8 wave32s) on one WGP; all intermediates live in LDS.
//  - Matrix math: V_WMMA_F32_16X16X4_F32 (the CDNA5 fp32 WMMA shape)
//  - MU staged to LDS by the Tensor Data Mover (zero-padded rows via OOB
//    semantics, bank-conflict-free pitch via TDM LDS padding)
// ---------------------------------------------------------------------------

typedef __attribute__((ext_vector_type(2))) float v2f;
typedef __attribute__((ext_vector_type(8))) float v8f;
typedef unsigned int __attribute__((ext_vector_type(4))) u32x4;
typedef int  __attribute__((ext_vector_type(4))) i32x4;
typedef int  __attribute__((ext_vector_type(8))) i32x8;

#define WAVES    8
#define THREADS  (WAVES * 32)

#define Bsz 32
#define Dd  128
#define Kk  512
#define Cc  10

// LDS pitches chosen to avoid bank conflicts on strided reads.
#define F2_PITCH 516   // fac2 row pitch (516 % 64 == 4 -> 16 lanes, distinct banks)
#define MU_PITCH 132   // 128 + 4 pad DWORDs inserted by TDM (132 % 64 == 4)
#define ST_PITCH 17    // staging tile pitch (odd -> conflict-free)

static __device__ __forceinline__ v8f wmma_f32(v2f a, v2f b, v8f c) {
    return __builtin_amdgcn_wmma_f32_16x16x4_f32(
        /*neg_a=*/false, a, /*neg_b=*/false, b,
        /*c_mod=*/(short)0, c, /*reuse_a=*/false, /*reuse_b=*/false);
}

// Compute one 16x16 fac1 tile (rows mt*16.., cols nt*16..) with 32 WMMA
// k-steps over D=128, stage it through LDS to convert C/D layout into
// A-fragment layout, then fold into the running h accumulator with 4 more
// WMMA k-steps against fac2^T.
static __device__ __forceinline__ v8f fac1_tile_and_h(
    int mt, int nt, v8f hacc,
    const float* __restrict__ X, const float* __restrict__ U,
    const float* __restrict__ fac2_s, float* __restrict__ stg,
    int l15, int hi)
{
    const int kb2 = 2 * hi;              // this half-wave's K offset within a 4-step
    const int m   = mt * 16 + l15;       // A row for this lane
    const int n   = nt * 16 + l15;       // B column for this lane

    v8f acc = {};
#pragma unroll
    for (int ks = 0; ks < Dd / 4; ++ks) {
        const int kb = ks * 4 + kb2;
        // X row fragment: 2 adjacent f32 -> one 8-byte load (kb is even)
        const v2f a = *reinterpret_cast<const v2f*>(&X[m * Dd + kb]);
        v2f b;
        b.x = U[kb * Kk + n];
        b.y = U[(kb + 1) * Kk + n];
        acc = wmma_f32(a, b, acc);
    }

    // Stage D tile (VGPR r holds row r + 8*hi, col = l15) into per-wave LDS.
    // DS ops from one wave are in-order; compiler inserts s_wait_dscnt.
#pragma unroll
    for (int r = 0; r < 8; ++r)
        stg[(r + 8 * hi) * ST_PITCH + l15] = acc[r];

    // h partial: A = fac1 tile [16x16], B = fac2_tile^T (B[k][c] = fac2[c][k]).
#pragma unroll
    for (int ks = 0; ks < 4; ++ks) {
        const int kb = ks * 4 + kb2;       // k within this 16-wide tile
        const int kg = nt * 16 + kb;       // global k into fac2
        v2f a, b;
        a.x = stg[l15 * ST_PITCH + kb];
        a.y = stg[l15 * ST_PITCH + kb + 1];
        b.x = fac2_s[l15 * F2_PITCH + kg];
        b.y = fac2_s[l15 * F2_PITCH + kg + 1];
        hacc = wmma_f32(a, b, hacc);
    }
    return hacc;
}

__global__ __launch_bounds__(THREADS)
void hsim_fused_kernel(const float* __restrict__ X,    // [32,128]
                       const float* __restrict__ MU,   // [10,128]
                       const float* __restrict__ U,    // [128,512]
                       const float* __restrict__ V,    // [128,512]
                       float* __restrict__ out)        // [32,10]
{
    __shared__ float fac2_s[16 * F2_PITCH];            // ~33 KB
    __shared__ float mu_s[16 * MU_PITCH];              // ~8.4 KB (TDM destination)
    __shared__ float stage_s[WAVES][16 * ST_PITCH];    // ~8.7 KB
    __shared__ float h_s[Bsz * 16];                    // 2 KB

    const int tid  = threadIdx.x;
    const int lane = tid & 31;
    const int wave = tid >> 5;
    const int l15  = lane & 15;
    const int hi   = lane >> 4;     // 0: lanes 0-15, 1: lanes 16-31
    const int kb2  = 2 * hi;

    // Zero the cross-wave h accumulator.
    for (int i = tid; i < Bsz * 16; i += THREADS) h_s[i] = 0.0f;

    // ---- Phase 0: stage MU[10,128] -> mu_s[16,132] (rows 10..15 zero).
#if defined(__has_builtin) && __has_builtin(__builtin_amdgcn_tensor_load_to_lds)
    if (wave == 0) {   // TDM ignores EXEC; issue from one wave only
        const unsigned long long ga = (unsigned long long)(uintptr_t)MU;
        const unsigned lds_addr    = (unsigned)(uintptr_t)&mu_s[0];

        // D# group 0 (ISA 8.3): count=1, lds_addr, global_addr, type=2
        u32x4 g0;
        g0[0] = 1u;                                   // count=1, user descriptor
        g0[1] = lds_addr;                             // LDS byte address
        g0[2] = (unsigned)(ga & 0xFFFFFFFFull);       // global_addr[31:0]
        g0[3] = (unsigned)((ga >> 32) & 0x1FFFFFFull) // global_addr[56:32]
              | (2u << 30);                           // type=2 ("image")

        // D# group 1 (ISA 8.4): data_size=4B; pad 4 DWORDs every 128 DWORDs
        // (-> LDS pitch 132); tensor 128x10, tile 128x16 (rows 10..15 OOB -> 0)
        i32x8 g1 = {};
        g1[0] = (2 << 16)        // data_size = 2 (4 bytes)
              | (1 << 20)        // pad_enable
              | (6 << 22)        // pad_interval: 2^(6+1) = 128 DWORDs
              | (3 << 25);       // pad_amount: 3+1 = 4 DWORDs
        g1[1] = (Dd & 0xFFFF) << 16;           // tensor_dim0[15:0] = 128
        g1[2] = (Cc & 0xFFFF) << 16;           // tensor_dim0[31:16]=0 | tensor_dim1 = 10
        g1[3] = (Dd & 0xFFFF) << 16;           // tensor_dim1[31:16]=0 | tile_dim0 = 128
        g1[4] = 16;                            // tile_dim1 = 16, tile_dim2 = 0
        g1[5] = Dd;                            // tensor_dim0_stride[31:0] = 128
        g1[6] = 0;                             // stride0 hi | stride1 lo
        g1[7] = 0;                             // stride1 hi

        i32x4 g2 = {};                         // 2D tensor: groups 2/3 unused
        i32x4 g3 = {};
#if __clang_major__ >= 23
        i32x8 g4 = {};
        __builtin_amdgcn_tensor_load_to_lds(g0, g1, g2, g3, g4, 0);
#else
        __builtin_amdgcn_tensor_load_to_lds(g0, g1, g2, g3, 0);
#endif
        __builtin_amdgcn_s_wait_tensorcnt(0);
    }
#else
    // Fallback: plain staged copy with zero padding.
    for (int i = tid; i < 16 * Dd; i += THREADS) {
        const int r = i >> 7, c = i & 127;
        mu_s[r * MU_PITCH + c] = (r < Cc) ? MU[r * Dd + c] : 0.0f;
    }
#endif
    __syncthreads();

    // ---- Phase 1: fac2 = MU @ V, 16-column tiles round-robined over waves.
    for (int nt = wave; nt < Kk / 16; nt += WAVES) {
        const int m = l15;
        const int n = nt * 16 + l15;
        v8f acc = {};
#pragma unroll
        for (int ks = 0; ks < Dd / 4; ++ks) {
            const int kb = ks * 4 + kb2;
            const v2f a = *reinterpret_cast<const v2f*>(&mu_s[m * MU_PITCH + kb]);
            v2f b;
            b.x = V[kb * Kk + n];
            b.y = V[(kb + 1) * Kk + n];
            acc = wmma_f32(a, b, acc);
        }
#pragma unroll
        for (int r = 0; r < 8; ++r)
            fac2_s[(r + 8 * hi) * F2_PITCH + n] = acc[r];
    }
    __syncthreads();

    // ---- Phase 2: per owned N-tile, build both fac1 M-tiles and fold them
    // straight into per-wave h partials (fac1 never touches global memory).
    v8f hacc0 = {};   // h rows 0..15
    v8f hacc1 = {};   // h rows 16..31
    float* stg = &stage_s[wave][0];
    for (int nt = wave; nt < Kk / 16; nt += WAVES) {
        if (nt + WAVES < Kk / 16)   // hint next U tile into cache
            __builtin_prefetch(&U[(nt + WAVES) * 16 + l15], 0, 0);
        hacc0 = fac1_tile_and_h(0, nt, hacc0, X, U, fac2_s, stg, l15, hi);
        hacc1 = fac1_tile_and_h(1, nt, hacc1, X, U, fac2_s, stg, l15, hi);
    }

    // Cross-wave reduction of h partials via LDS float atomics (ds_add_f32).
#pragma unroll
    for (int r = 0; r < 8; ++r) {
        atomicAdd(&h_s[(r + 8 * hi) * 16 + l15],      hacc0[r]);
        atomicAdd(&h_s[(16 + r + 8 * hi) * 16 + l15], hacc1[r]);
    }
    __syncthreads();

    // ---- Phase 3: row-wise log_softmax over the 10 valid classes.
    if (tid < Bsz) {
        float mx = -INFINITY;
#pragma unroll
        for (int c = 0; c < Cc; ++c) mx = fmaxf(mx, h_s[tid * 16 + c]);
        float s = 0.0f;
#pragma unroll
        for (int c = 0; c < Cc; ++c) s += __expf(h_s[tid * 16 + c] - mx);
        const float lse = mx + __logf(s);
#pragma unroll
        for (int c = 0; c < Cc; ++c) out[tid * Cc + c] = h_s[tid * 16 + c] - lse;
    }
}

extern "C" void kernel_launch(void* const* d_in, const int* in_sizes, int n_in,
                              void* d_out, int out_size, void* d_ws, size_t ws_size,
                              hipStream_t stream) {
    (void)in_sizes; (void)n_in; (void)d_ws; (void)ws_size; (void)out_size;
    const float* X  = (const float*)d_in[0];   // x_batch [32,128]
    const float* MU = (const float*)d_in[1];   // mu_c    [10,128]
    const float* U  = (const float*)d_in[2];   // U       [128,512]
    const float* V  = (const float*)d_in[3];   // V       [128,512]
    float* out = (float*)d_out;                // [32,10]

    hsim_fused_kernel<<<1, THREADS, 0, stream>>>(X, MU, U, V, out);
}